// GATWithEmbedding_36283883717209
// MI455X (gfx1250) — compile-verified
//
#include <hip/hip_runtime.h>
#include <hip/hip_bf16.h>
#include <math.h>

#define HEADS 4
#define CHAN 32
#define FEAT 128
#define EMB_IN 32
#define NEG_SLOPE 0.2f

typedef __attribute__((ext_vector_type(2))) float v2f;
typedef __attribute__((ext_vector_type(8))) float v8f;

__device__ __forceinline__ float lrelu(float x) { return x > 0.f ? x : NEG_SLOPE * x; }

// order-preserving float<->uint encoding for atomicMax on floats (handles negatives)
__device__ __forceinline__ unsigned f2ord(float f) {
    unsigned u = __float_as_uint(f);
    return (u & 0x80000000u) ? ~u : (u | 0x80000000u);
}
__device__ __forceinline__ float ord2f(unsigned e) {
    unsigned u = (e & 0x80000000u) ? (e ^ 0x80000000u) : ~e;
    return __uint_as_float(u);
}
#define ORD_NEG_INF 0x007FFFFFu  // f2ord(-inf)

__device__ __forceinline__ void edge_sd(const long long* __restrict__ ei, long long E,
                                        long long e, int& s, int& d) {
    if (e < E) { s = (int)ei[e]; d = (int)ei[E + e]; }
    else       { s = d = (int)(e - E); }   // self-loop
}

// ---------------- gather: h0[n,c] = emb[x[n], c] ----------------
__global__ void gather_kernel(const long long* __restrict__ x, const float* __restrict__ emb,
                              float* __restrict__ h0, int n) {
    int t = blockIdx.x * blockDim.x + threadIdx.x;
    if (t >= n * EMB_IN) return;
    int node = t >> 5, c = t & 31;
    long long src = x[node];
    h0[t] = emb[src * EMB_IN + c];
}

// ---------------- GEMM: H[N,128] = X[N,K] @ W[128,K]^T via V_WMMA_F32_16X16X4_F32 ----
// block = 256 threads (8 waves). The 16xK A tile is staged once into LDS (padded
// stride K+1 against bank conflicts); wave w computes the 16x16 tile at cols
// [16w,16w+16). Full-tile blocks take an unguarded store fast path.
template <int K>
__global__ void gemm_wmma_kernel(const float* __restrict__ X, const float* __restrict__ W,
                                 float* __restrict__ H, int n) {
    __shared__ float sA[16][K + 1];

    const int wave = threadIdx.x >> 5;
    const int lane = threadIdx.x & 31;
    const int rowBase = blockIdx.x * 16;

    // cooperative A-tile stage: 16*K elements, exact multiple of 256 threads
#pragma unroll
    for (int i = threadIdx.x; i < 16 * K; i += 256) {
        int rr = rowBase + i / K;
        if (rr >= n) rr = n - 1;                       // clamp, no divergence
        sA[i / K][i % K] = X[(size_t)rr * K + (i % K)];
    }
    __syncthreads();

    const int col = wave * 16 + (lane & 15);
    const int arow = lane & 15;
    const int kh = (lane >> 4) << 1;                   // lanes 0-15 -> K{0,1}; 16-31 -> K{2,3}
    const float* wr = W + (size_t)col * K;

    v8f acc = {0.f, 0.f, 0.f, 0.f, 0.f, 0.f, 0.f, 0.f};
#pragma unroll
    for (int k0 = 0; k0 < K; k0 += 4) {
        v2f a, b;
        a.x = sA[arow][k0 + kh]; a.y = sA[arow][k0 + kh + 1];   // ds_load_b64
        b.x = wr[k0 + kh];       b.y = wr[k0 + kh + 1];         // global_load_b64
#if __has_builtin(__builtin_amdgcn_wmma_f32_16x16x4_f32)
        acc = __builtin_amdgcn_wmma_f32_16x16x4_f32(false, a, false, b, (short)0, acc,
                                                    false, false);
#else
        acc[0] += a.x * b.x + a.y * b.y;               // placeholder for host pass only
#endif
    }

    const int m0 = (lane >> 4) * 8;                    // C/D layout: vgpr v -> M = v + 8*(lane>=16)
    if (rowBase + 16 <= n) {                           // uniform fast path: no lane guards
#pragma unroll
        for (int v = 0; v < 8; ++v)
            H[(size_t)(rowBase + m0 + v) * FEAT + col] = acc[v];
    } else {
#pragma unroll
        for (int v = 0; v < 8; ++v) {
            int r = rowBase + m0 + v;
            if (r < n) H[(size_t)r * FEAT + col] = acc[v];
        }
    }
}

// ---------------- per-node attention logits: a_src[n,h], a_dst[n,h] ----------------
__global__ void att_kernel(const float* __restrict__ H, const float* __restrict__ att_src,
                           const float* __restrict__ att_dst, float* __restrict__ asrc,
                           float* __restrict__ adst, int n) {
    int t = blockIdx.x * blockDim.x + threadIdx.x;
    if (t >= n * HEADS) return;
    int node = t >> 2, h = t & 3;
    const float* hr = H + (size_t)node * FEAT + h * CHAN;
    const float* as = att_src + h * CHAN;
    const float* ad = att_dst + h * CHAN;
    float s = 0.f, d = 0.f;
#pragma unroll
    for (int c = 0; c < CHAN; ++c) { float v = hr[c]; s += v * as[c]; d += v * ad[c]; }
    asrc[t] = s; adst[t] = d;
}

// ---------------- init segment-softmax state ----------------
__global__ void init_seg_kernel(unsigned* __restrict__ amax, float* __restrict__ denom, int n) {
    int t = blockIdx.x * blockDim.x + threadIdx.x;
    if (t >= n * HEADS) return;
    amax[t] = ORD_NEG_INF;
    denom[t] = 0.f;
}
__global__ void init_acc_kernel(float* __restrict__ acc, const float* __restrict__ bias, int n) {
    int t = blockIdx.x * blockDim.x + threadIdx.x;
    if (t >= n * FEAT) return;
    acc[t] = bias[t & (FEAT - 1)];
}

// ---------------- edge pass 1: segment max ----------------
__global__ void edge_max_kernel(const long long* __restrict__ ei, long long E, long long ET,
                                const float* __restrict__ asrc, const float* __restrict__ adst,
                                unsigned* __restrict__ amax) {
    long long e = (long long)blockIdx.x * blockDim.x + threadIdx.x;
    if (e >= ET) return;
    int s, d; edge_sd(ei, E, e, s, d);
#pragma unroll
    for (int h = 0; h < HEADS; ++h) {
        float a = lrelu(asrc[s * HEADS + h] + adst[d * HEADS + h]);
        atomicMax(&amax[d * HEADS + h], f2ord(a));
    }
}

// ---------------- edge pass 2: segment sum of exp ----------------
__global__ void edge_sum_kernel(const long long* __restrict__ ei, long long E, long long ET,
                                const float* __restrict__ asrc, const float* __restrict__ adst,
                                const unsigned* __restrict__ amax, float* __restrict__ denom) {
    long long e = (long long)blockIdx.x * blockDim.x + threadIdx.x;
    if (e >= ET) return;
    int s, d; edge_sd(ei, E, e, s, d);
#pragma unroll
    for (int h = 0; h < HEADS; ++h) {
        float a = lrelu(asrc[s * HEADS + h] + adst[d * HEADS + h]);
        float ex = __expf(a - ord2f(amax[d * HEADS + h]));
        atomicAdd(&denom[d * HEADS + h], ex);
    }
}

// ---------------- edge pass 3: weighted scatter. wave-per-edge, lane = 4 channels ---
__global__ void edge_aggr_kernel(const long long* __restrict__ ei, long long E, long long ET,
                                 const float* __restrict__ asrc, const float* __restrict__ adst,
                                 const unsigned* __restrict__ amax,
                                 const float* __restrict__ denom,
                                 const float* __restrict__ H, float* __restrict__ acc) {
    long long e = (long long)blockIdx.x * (blockDim.x >> 5) + (threadIdx.x >> 5);
    if (e >= ET) return;
    const int lane = threadIdx.x & 31;
    int s, d; edge_sd(ei, E, e, s, d);

    const float* hs = H + (size_t)s * FEAT;
    __builtin_prefetch(hs + lane * 4, 0, 0);   // global_prefetch_b8 on the src row

    const int h = lane >> 3;                   // 8 lanes (32 channels) per head
    float a = lrelu(asrc[s * HEADS + h] + adst[d * HEADS + h]);
    float m = ord2f(amax[d * HEADS + h]);
    float coef = __expf(a - m) / (denom[d * HEADS + h] + 1e-16f);

    const int c0 = lane * 4;
    float4 hv = *(const float4*)(hs + c0);     // global_load_b128
    float* ap = acc + (size_t)d * FEAT + c0;
    atomicAdd(ap + 0, coef * hv.x);
    atomicAdd(ap + 1, coef * hv.y);
    atomicAdd(ap + 2, coef * hv.z);
    atomicAdd(ap + 3, coef * hv.w);
}

__global__ void relu_kernel(float* __restrict__ a, int n) {
    int t = blockIdx.x * blockDim.x + threadIdx.x;
    if (t >= n * FEAT) return;
    float v = a[t];
    a[t] = v > 0.f ? v : 0.f;
}

extern "C" void kernel_launch(void* const* d_in, const int* in_sizes, int n_in,
                              void* d_out, int out_size, void* d_ws, size_t ws_size,
                              hipStream_t stream) {
    const long long* x  = (const long long*)d_in[0];
    const long long* ei = (const long long*)d_in[1];
    const float* emb = (const float*)d_in[2];
    const float* W1  = (const float*)d_in[3];
    const float* as1 = (const float*)d_in[4];
    const float* ad1 = (const float*)d_in[5];
    const float* b1  = (const float*)d_in[6];
    const float* W2  = (const float*)d_in[7];
    const float* as2 = (const float*)d_in[8];
    const float* ad2 = (const float*)d_in[9];
    const float* b2  = (const float*)d_in[10];

    const int N = in_sizes[0];
    const long long E  = (long long)in_sizes[1] / 2;
    const long long ET = E + N;

    char* ws = (char*)d_ws;
    size_t off = 0;
    auto walloc = [&](size_t bytes) -> void* {
        void* p = ws + off;
        off += (bytes + 255) & ~(size_t)255;
        return p;
    };
    float*    h0    = (float*)walloc((size_t)N * EMB_IN * sizeof(float));
    float*    Hf    = (float*)walloc((size_t)N * FEAT * sizeof(float));
    float*    acc1  = (float*)walloc((size_t)N * FEAT * sizeof(float));
    float*    asrc  = (float*)walloc((size_t)N * HEADS * sizeof(float));
    float*    adst  = (float*)walloc((size_t)N * HEADS * sizeof(float));
    unsigned* amax  = (unsigned*)walloc((size_t)N * HEADS * sizeof(unsigned));
    float*    denom = (float*)walloc((size_t)N * HEADS * sizeof(float));
    float*    out   = (float*)d_out;

    const dim3 blk(256);
    const unsigned gNH  = (unsigned)((N * HEADS + 255) / 256);
    const unsigned gNF  = (unsigned)(((size_t)N * FEAT + 255) / 256);
    const unsigned gRow = (unsigned)((N + 15) / 16);
    const unsigned gE   = (unsigned)((ET + 255) / 256);
    const unsigned gE8  = (unsigned)((ET + 7) / 8);      // 8 waves per block, wave-per-edge

    // ---------- layer 1 ----------
    gather_kernel<<<(unsigned)((N * EMB_IN + 255) / 256), blk, 0, stream>>>(x, emb, h0, N);
    gemm_wmma_kernel<EMB_IN><<<gRow, blk, 0, stream>>>(h0, W1, Hf, N);
    att_kernel<<<gNH, blk, 0, stream>>>(Hf, as1, ad1, asrc, adst, N);
    init_seg_kernel<<<gNH, blk, 0, stream>>>(amax, denom, N);
    init_acc_kernel<<<gNF, blk, 0, stream>>>(acc1, b1, N);
    edge_max_kernel<<<gE, blk, 0, stream>>>(ei, E, ET, asrc, adst, amax);
    edge_sum_kernel<<<gE, blk, 0, stream>>>(ei, E, ET, asrc, adst, amax, denom);
    edge_aggr_kernel<<<gE8, blk, 0, stream>>>(ei, E, ET, asrc, adst, amax, denom, Hf, acc1);
    relu_kernel<<<gNF, blk, 0, stream>>>(acc1, N);

    // ---------- layer 2 ----------
    gemm_wmma_kernel<FEAT><<<gRow, blk, 0, stream>>>(acc1, W2, Hf, N);
    att_kernel<<<gNH, blk, 0, stream>>>(Hf, as2, ad2, asrc, adst, N);
    init_seg_kernel<<<gNH, blk, 0, stream>>>(amax, denom, N);
    init_acc_kernel<<<gNF, blk, 0, stream>>>(out, b2, N);
    edge_max_kernel<<<gE, blk, 0, stream>>>(ei, E, ET, asrc, adst, amax);
    edge_sum_kernel<<<gE, blk, 0, stream>>>(ei, E, ET, asrc, adst, amax, denom);
    edge_aggr_kernel<<<gE8, blk, 0, stream>>>(ei, E, ET, asrc, adst, amax, denom, Hf, out);
}